// MatrixFactorization_19370302505036
// MI455X (gfx1250) — compile-verified
//
#include <hip/hip_runtime.h>

// MI455X / gfx1250, wave32. out[b] = dot(user_factors[uidx[b]], s),
// s = sum_c item_factors[iidx[c]]. Bandwidth-bound (~8MB gathered @ 23.3TB/s).
// Dot phase runs on the matrix pipe via V_WMMA_F32_16X16X4_F32 (f32-exact).

typedef __attribute__((ext_vector_type(2))) float v2f;
typedef __attribute__((ext_vector_type(8))) float v8f;

#define FDIM 64

// Phase 1: per-block partial sums of gathered item rows (deterministic tree).
__global__ __launch_bounds__(256) void mf_item_partial(
    const int* __restrict__ item_idx,
    const float* __restrict__ item_factors,
    float* __restrict__ partials,      // [gridDim.x][FDIM]
    int per_block)
{
    __shared__ float4 sh[256];
    const int t = threadIdx.x;
    const int r = t >> 4;              // 0..15: row slot within tile
    const int q = t & 15;              // 0..15: float4 slice of the 64-f row
    const int base = blockIdx.x * per_block;

    float4 acc = make_float4(0.f, 0.f, 0.f, 0.f);
    const int iters = per_block >> 4;
    for (int i = 0; i < iters; ++i) {
        const int c = base + i * 16 + r;
        const long long row = (long long)item_idx[c] * FDIM;
        const float4 v = *((const float4*)(item_factors + row) + q);
        acc.x += v.x; acc.y += v.y; acc.z += v.z; acc.w += v.w;
    }
    sh[t] = acc;
    __syncthreads();
    #pragma unroll
    for (int s = 8; s >= 1; s >>= 1) {
        if (r < s) {
            float4 o = sh[(r + s) * 16 + q];
            float4 m = sh[t];
            m.x += o.x; m.y += o.y; m.z += o.z; m.w += o.w;
            sh[t] = m;
        }
        __syncthreads();
    }
    if (r == 0) {
        float* dst = partials + (size_t)blockIdx.x * FDIM + q * 4;
        const float4 v = sh[q];
        dst[0] = v.x; dst[1] = v.y; dst[2] = v.z; dst[3] = v.w;
    }
}

// Phase 2: fold partials -> s (LDS), then per-wave 16-row GEMV on WMMA.
// A(16x4) = user rows slice; B(4x16) = s chunk broadcast across columns.
// f32 A/B 32-bit operand striping: lane<16 holds K=kb+0,kb+1; lane>=16 holds
// K=kb+2,kb+3 => identical float2 addressing for A and B via koff.
__global__ __launch_bounds__(256) void mf_gemv_wmma(
    const int* __restrict__ user_idx,
    const float* __restrict__ user_factors,
    const float* __restrict__ partials,
    float* __restrict__ out,
    int n_partials)
{
    __shared__ float s_sh[FDIM];
    const int t = threadIdx.x;
    if (t < FDIM) {
        float sum = 0.f;
        for (int g = 0; g < n_partials; ++g)     // fixed order: deterministic
            sum += partials[(size_t)g * FDIM + t];
        s_sh[t] = sum;
    }
    __syncthreads();

    const int lane = t & 31;
    const int wave = t >> 5;
    const int m    = lane & 15;
    const int koff = (lane < 16) ? 0 : 2;
    const int rowbase = (blockIdx.x * 8 + wave) * 16;

    const long long urow = (long long)user_idx[rowbase + m] * FDIM;
    const float* up = user_factors + urow;       // 256B contiguous per row

    v8f c = {};
    #pragma unroll
    for (int kk = 0; kk < 16; ++kk) {
        const int kb = kk * 4;
        const v2f a = *(const v2f*)(up + kb + koff);
        const v2f b = *(const v2f*)(&s_sh[kb + koff]);
        c = __builtin_amdgcn_wmma_f32_16x16x4_f32(
                /*neg_a=*/false, a, /*neg_b=*/false, b,
                /*c_mod=*/(short)0, c, /*reuse_a=*/false, /*reuse_b=*/false);
    }

    // D: VGPR r, lanes 0-15 -> (M=r, N=lane); lanes 16-31 -> (M=r+8).
    // All columns equal, so lanes 0 and 16 write rows 0-7 and 8-15.
    if ((lane & 15) == 0) {
        const int mb = rowbase + ((lane >> 4) << 3);
        #pragma unroll
        for (int r2 = 0; r2 < 8; ++r2) out[mb + r2] = c[r2];
    }
}

extern "C" void kernel_launch(void* const* d_in, const int* in_sizes, int n_in,
                              void* d_out, int out_size, void* d_ws, size_t ws_size,
                              hipStream_t stream) {
    const int*   user_idx     = (const int*)d_in[0];
    const int*   item_idx     = (const int*)d_in[1];
    const float* user_factors = (const float*)d_in[2];
    const float* item_factors = (const float*)d_in[3];
    float*       out          = (float*)d_out;
    float*       partials     = (float*)d_ws;

    const int B = in_sizes[0];          // 16384
    const int per_block = 256;
    const int g1 = B / per_block;       // 64 partial blocks -> 16 KB in d_ws

    mf_item_partial<<<g1, 256, 0, stream>>>(item_idx, item_factors, partials, per_block);

    const int g2 = B / 128;             // 8 waves/block * 16 rows/wave
    mf_gemv_wmma<<<g2, 256, 0, stream>>>(user_idx, user_factors, partials, out, g1);
}